// TraceFriendlyLinearAttention_45208825758214
// MI455X (gfx1250) — compile-verified
//
#include <hip/hip_runtime.h>

typedef float v2f __attribute__((ext_vector_type(2)));
typedef float v8f __attribute__((ext_vector_type(8)));

#define BDIM 2
#define HDIM 36
#define DDIM 32
#define NDIM 4096
#define BH   (BDIM * HDIM)
#define SROWS 33
#define SSZ  (SROWS * DDIM)   /* 1056 floats per (b,h) score matrix incl. normalizer row */

#define NSPLIT 8              /* N-dimension slices for the scores reduction */
#define SLICE_N (NDIM / NSPLIT)

#define WMMA_F32(A, B, C) \
  __builtin_amdgcn_wmma_f32_16x16x4_f32(false, (A), false, (B), (short)0, (C), false, false)

// ---- CDNA5 async global->LDS copies (ASYNCcnt path, no data VGPRs) --------
__device__ __forceinline__ unsigned lds_off(const void* p) {
  // Generic pointers to LDS are {shared_aperture_hi32, lds_offset32}.
  return (unsigned)(uintptr_t)p;
}
__device__ __forceinline__ void async_b128(unsigned lds, unsigned long long ga) {
  asm volatile("global_load_async_to_lds_b128 %0, %1, off" :: "v"(lds), "v"(ga) : "memory");
}
__device__ __forceinline__ void async_b32(unsigned lds, unsigned long long ga) {
  asm volatile("global_load_async_to_lds_b32 %0, %1, off" :: "v"(lds), "v"(ga) : "memory");
}
__device__ __forceinline__ void wait_async0() {
  asm volatile("s_wait_asynccnt 0" ::: "memory");
}

// ---------------------------------------------------------------------------
// Kernel 1: partial S[d][e] = sum_{n in slice} Vpad[d][n] * K[e][n]
// Grid (NSPLIT, BH) = 576 blocks. Double-buffered async staging: 64-token
// chunks, stride 68 (bank = 4*row + col -> conflict-free b64 gathers, rows
// 16B-aligned). One barrier per chunk; async DMA overlaps WMMA.
// ---------------------------------------------------------------------------
#define K1_BT 64
#define K1_ST 68
#define K1_BUF (2 * 32 * K1_ST)     /* 4352 floats per buffer (V + K) */
#define CHUNKS (SLICE_N / K1_BT)    /* 8 */

__global__ __launch_bounds__(256)
void la_scores_kernel(const float* __restrict__ key,
                      const float* __restrict__ value,
                      float* __restrict__ S_part) {
  __shared__ float smem[2 * K1_BUF];   // 8704 floats; reused for reduction (8*1056=8448)

  const int slice = blockIdx.x;
  const int bh    = blockIdx.y;
  const float* kp = key   + (size_t)bh * DDIM * NDIM;
  const float* vp = value + (size_t)bh * DDIM * NDIM;
  const int nbeg  = slice * SLICE_N;

  const int tid  = threadIdx.x;
  const int w    = tid >> 5;        // wave id 0..7 (uniform per wave)
  const int lane = tid & 31;
  const int half = lane >> 4;       // 0 | 1
  const int m    = lane & 15;

  const int srow = tid >> 4;        // staging: row 0..15 (two passes of 16 rows)
  const int scol = (tid & 15) * 4;  // staging: float4 column

  v8f acc00 = {0.f,0.f,0.f,0.f,0.f,0.f,0.f,0.f};
  v8f acc01 = acc00, acc10 = acc00, acc11 = acc00, acc20 = acc00, acc21 = acc00;

  // Synthetic "ones" row tile for the normalizer (global row d=32): A[m][k]=1 iff m==0.
  v2f a2;
  a2.x = (m == 0) ? 1.0f : 0.0f;
  a2.y = a2.x;

  // Prologue: issue chunk 0 into buffer 0.
  {
    float* bV = smem;
    float* bK = smem + 32 * K1_ST;
#pragma unroll
    for (int p = 0; p < 2; ++p) {
      const int row = srow + p * 16;
      async_b128(lds_off(&bV[row * K1_ST + scol]),
                 (unsigned long long)(vp + (size_t)row * NDIM + nbeg + scol));
      async_b128(lds_off(&bK[row * K1_ST + scol]),
                 (unsigned long long)(kp + (size_t)row * NDIM + nbeg + scol));
    }
  }

  for (int c = 0; c < CHUNKS; ++c) {
    wait_async0();       // my chunk-c transfers landed in LDS
    __syncthreads();     // everyone's landed; previous compute done with other buffer

    if (c + 1 < CHUNKS) {   // prefetch next chunk while we compute on this one
      float* nb = smem + ((c + 1) & 1) * K1_BUF;
      float* bV = nb;
      float* bK = nb + 32 * K1_ST;
      const int t0 = nbeg + (c + 1) * K1_BT;
#pragma unroll
      for (int p = 0; p < 2; ++p) {
        const int row = srow + p * 16;
        async_b128(lds_off(&bV[row * K1_ST + scol]),
                   (unsigned long long)(vp + (size_t)row * NDIM + t0 + scol));
        async_b128(lds_off(&bK[row * K1_ST + scol]),
                   (unsigned long long)(kp + (size_t)row * NDIM + t0 + scol));
      }
    }

    const float* sV = smem + (c & 1) * K1_BUF;
    const float* sK = sV + 32 * K1_ST;
    const int cbase = w * 8;
#pragma unroll
    for (int s = 0; s < 2; ++s) {
      const int col = cbase + s * 4 + 2 * half;       // k = 2*half + v within this step
      const v2f a0 = *(const v2f*)(&sV[m        * K1_ST + col]);
      const v2f a1 = *(const v2f*)(&sV[(m + 16) * K1_ST + col]);
      const v2f b0 = *(const v2f*)(&sK[m        * K1_ST + col]);
      const v2f b1 = *(const v2f*)(&sK[(m + 16) * K1_ST + col]);
      acc00 = WMMA_F32(a0, b0, acc00);
      acc01 = WMMA_F32(a0, b1, acc01);
      acc10 = WMMA_F32(a1, b0, acc10);
      acc11 = WMMA_F32(a1, b1, acc11);
      acc20 = WMMA_F32(a2, b0, acc20);
      acc21 = WMMA_F32(a2, b1, acc21);
    }
  }
  __syncthreads();   // all compute done before reusing LDS for the reduction

  // Per-wave partial S -> LDS, D layout: (M = r + 8*half, N = m).
  float* mine = smem + w * SSZ;
#pragma unroll
  for (int r = 0; r < 8; ++r) {
    const int d0 = r + 8 * half;
    mine[d0        * 32 + m     ] = acc00[r];
    mine[d0        * 32 + 16 + m] = acc01[r];
    mine[(d0 + 16) * 32 + m     ] = acc10[r];
    mine[(d0 + 16) * 32 + 16 + m] = acc11[r];
  }
  if (lane < 16) {                        // normalizer row: M=0 -> r=0, lanes 0..15
    mine[32 * 32 + lane     ] = acc20[0];
    mine[32 * 32 + 16 + lane] = acc21[0];
  }
  __syncthreads();

  float* dst = S_part + ((size_t)bh * NSPLIT + slice) * SSZ;
  for (int i = tid; i < SSZ; i += 256) {
    float s = 0.f;
#pragma unroll
    for (int ww = 0; ww < 8; ++ww) s += smem[ww * SSZ + i];
    dst[i] = s;
  }
}

// ---------------------------------------------------------------------------
// Kernel 1b: reduce the NSPLIT partial score matrices (fixed order -> exact
// determinism). Tiny: 72 blocks, ~2.7 MB of traffic.
// ---------------------------------------------------------------------------
__global__ __launch_bounds__(256)
void la_reduce_kernel(const float* __restrict__ S_part,
                      float* __restrict__ S_fin) {
  const int bh = blockIdx.x;
  const float* src = S_part + (size_t)bh * NSPLIT * SSZ;
  float* dst = S_fin + (size_t)bh * SSZ;
  for (int i = threadIdx.x; i < SSZ; i += 256) {
    float s = 0.f;
#pragma unroll
    for (int p = 0; p < NSPLIT; ++p) s += src[p * SSZ + i];
    dst[i] = s;
  }
}

// ---------------------------------------------------------------------------
// Kernel 2: H = S_pad(33x32) @ Q(32x4096); out[d][n] = H[d][n] / (H[32][n]+eps)
// Block = (token block of 256, bh). Async staging; Q stride 264 keeps the two
// lane halves of each B-gather on disjoint bank ranges.
// ---------------------------------------------------------------------------
#define K2_BT 256
#define QS    264
#define SST   34

__global__ __launch_bounds__(256)
void la_apply_kernel(const float* __restrict__ query,
                     const float* __restrict__ S_fin,
                     float* __restrict__ out) {
  __shared__ float sS[SROWS * SST];   // 1122 floats
  __shared__ float sQ[32 * QS];       // 8448 floats

  const int tb = blockIdx.x * K2_BT;
  const int bh = blockIdx.y;
  const float* qp = query + (size_t)bh * DDIM * NDIM;
  const float* Sp = S_fin + (size_t)bh * SSZ;

  const int tid  = threadIdx.x;
  const int w    = tid >> 5;
  const int lane = tid & 31;
  const int half = lane >> 4;
  const int m    = lane & 15;

  for (int i = tid; i < SSZ; i += 256)
    async_b32(lds_off(&sS[(i >> 5) * SST + (i & 31)]),
              (unsigned long long)(Sp + i));
  {
    const int col = (tid & 63) * 4;
    const int rb  = tid >> 6;           // 4 rows per pass
#pragma unroll
    for (int p = 0; p < 8; ++p) {
      const int row = rb + p * 4;
      async_b128(lds_off(&sQ[row * QS + col]),
                 (unsigned long long)(qp + (size_t)row * NDIM + tb + col));
    }
  }
  wait_async0();
  __syncthreads();

  float* ob = out + (size_t)bh * DDIM * NDIM;

#pragma unroll
  for (int tt = 0; tt < 2; ++tt) {
    const int t0 = w * 32 + tt * 16;

    v8f acc0 = {0.f,0.f,0.f,0.f,0.f,0.f,0.f,0.f};
    v8f acc1 = acc0, acc2 = acc0;

#pragma unroll
    for (int k0 = 0; k0 < 32; k0 += 4) {
      const int kk = k0 + 2 * half;
      const v2f a0  = *(const v2f*)(&sS[m        * SST + kk]);
      const v2f a1  = *(const v2f*)(&sS[(m + 16) * SST + kk]);
      const v2f s32 = *(const v2f*)(&sS[32       * SST + kk]);   // broadcast read
      v2f a2;
      a2.x = (m == 0) ? s32.x : 0.0f;   // pad rows 33..47 are zero
      a2.y = (m == 0) ? s32.y : 0.0f;
      v2f b;
      b.x = sQ[kk       * QS + t0 + m];
      b.y = sQ[(kk + 1) * QS + t0 + m];
      acc0 = WMMA_F32(a0, b, acc0);
      acc1 = WMMA_F32(a1, b, acc1);
      acc2 = WMMA_F32(a2, b, acc2);
    }

    // H[32][t0+j] sits in acc2[0] of lane j (j<16); broadcast to both halves.
    const float nrm = __shfl(acc2[0], m, 32);
    const float inv = 1.0f / (nrm + 1e-15f);

    const int gt = tb + t0 + m;
#pragma unroll
    for (int r = 0; r < 8; ++r) {
      const int d0 = r + 8 * half;
      ob[(size_t)d0        * NDIM + gt] = acc0[r] * inv;
      ob[(size_t)(d0 + 16) * NDIM + gt] = acc1[r] * inv;
    }
  }
}

extern "C" void kernel_launch(void* const* d_in, const int* in_sizes, int n_in,
                              void* d_out, int out_size, void* d_ws, size_t ws_size,
                              hipStream_t stream) {
  (void)in_sizes; (void)n_in; (void)out_size; (void)ws_size;
  const float* q = (const float*)d_in[0];
  const float* k = (const float*)d_in[1];
  const float* v = (const float*)d_in[2];
  float* outp = (float*)d_out;

  // ws layout: [BH][NSPLIT][SSZ] partials, then [BH][SSZ] final  (~2.7 MB total)
  float* S_part = (float*)d_ws;
  float* S_fin  = S_part + (size_t)BH * NSPLIT * SSZ;

  la_scores_kernel<<<dim3(NSPLIT, BH), 256, 0, stream>>>(k, v, S_part);
  la_reduce_kernel<<<BH, 256, 0, stream>>>(S_part, S_fin);
  la_apply_kernel<<<dim3(NDIM / K2_BT, BH), 256, 0, stream>>>(q, S_fin, outp);
}